// GCN_22857815949622
// MI455X (gfx1250) — compile-verified
//
#include <hip/hip_runtime.h>
#include <hip/hip_bf16.h>

#define N_NODES  50000
#define N_EDGES  800000
#define HDIM     128
#define NCLS     3
#define NLAYERS  3
#define NGRAPH   512
#define BN_EPS   1e-5f

typedef float v2f __attribute__((ext_vector_type(2)));
typedef float v8f __attribute__((ext_vector_type(8)));

// ---------------------------------------------------------------------------
// Degree / normalization precompute
// ---------------------------------------------------------------------------
__global__ void deg_init_k(float* __restrict__ deg) {
    int i = blockIdx.x * blockDim.x + threadIdx.x;
    if (i < N_NODES) deg[i] = 1.0f;            // self-loop contributes 1
}

__global__ void deg_edge_k(const int* __restrict__ dst, float* __restrict__ deg) {
    int e = blockIdx.x * blockDim.x + threadIdx.x;
    if (e < N_EDGES) unsafeAtomicAdd(&deg[dst[e]], 1.0f);
}

__global__ void dinv_k(float* __restrict__ deg) {
    int i = blockIdx.x * blockDim.x + threadIdx.x;
    if (i < N_NODES) deg[i] = rsqrtf(deg[i]);  // deg >= 1 always
}

__global__ void enorm_k(const int* __restrict__ src, const int* __restrict__ dst,
                        const float* __restrict__ dinv, float* __restrict__ en) {
    int e = blockIdx.x * blockDim.x + threadIdx.x;
    if (e < N_EDGES) en[e] = dinv[src[e]] * dinv[dst[e]];
}

// ---------------------------------------------------------------------------
// Dense GEMM  out[N,128] = A[N,128] @ W[128,128]  via V_WMMA_F32_16X16X4_F32
// One wave per 16x16 output tile; 8 waves per block cover the 8 column tiles.
// ---------------------------------------------------------------------------
__global__ void __launch_bounds__(256)
gemm_wmma_k(const float* __restrict__ A, const float* __restrict__ W,
            float* __restrict__ out) {
    const int lane = threadIdx.x & 31;
    const int wv   = threadIdx.x >> 5;          // 0..7 -> column tile
    const int m0   = blockIdx.x << 4;           // row tile base (N % 16 == 0)
    const int n0   = wv << 4;
    const int r16  = lane & 15;
    const int hi   = lane >> 4;                 // 0: K pair {0,1}; 1: K pair {2,3}

    v8f acc = {};
    // A layout (16x4 f32): lane<16 holds K=k0,k0+1 ; lane>=16 holds K=k0+2,k0+3
    const float* arow = A + (size_t)(m0 + r16) * HDIM + 2 * hi;
    // B layout (4x16 f32): N striped across lanes, same K split by lane half
    const float* bcol = W + n0 + r16;

#pragma unroll
    for (int k0 = 0; k0 < HDIM; k0 += 4) {
        const float2 av = *(const float2*)(arow + k0);
        v2f a, b;
        a.x = av.x;
        a.y = av.y;
        b.x = bcol[(size_t)(k0 + 2 * hi) * HDIM];
        b.y = bcol[(size_t)(k0 + 2 * hi + 1) * HDIM];
        acc = __builtin_amdgcn_wmma_f32_16x16x4_f32(
            /*neg_a=*/false, a, /*neg_b=*/false, b,
            /*c_mod=*/(short)0, acc, /*reuse_a=*/false, /*reuse_b=*/false);
    }

    // C/D layout: VGPR r -> M=r (lanes 0-15) / M=r+8 (lanes 16-31), N = lane&15
    float* orow = out + (size_t)(m0 + hi * 8) * HDIM + n0 + r16;
#pragma unroll
    for (int r = 0; r < 8; ++r)
        orow[(size_t)r * HDIM] = acc[r];
}

// ---------------------------------------------------------------------------
// Aggregation: self-loop init then edge scatter-add (wave per edge, float4/lane)
// ---------------------------------------------------------------------------
__global__ void selfloop_k(const float* __restrict__ hw, const float* __restrict__ dinv,
                           float* __restrict__ agg) {
    int t = blockIdx.x * blockDim.x + threadIdx.x;   // N*32 threads exactly
    int n = t >> 5, c = t & 31;
    float s = dinv[n]; s *= s;                        // self-loop norm
    const float4 v = *(const float4*)(hw + (size_t)n * HDIM + c * 4);
    float4 o = make_float4(v.x * s, v.y * s, v.z * s, v.w * s);
    *(float4*)(agg + (size_t)n * HDIM + c * 4) = o;
}

__global__ void scatter_k(const int* __restrict__ src, const int* __restrict__ dst,
                          const float* __restrict__ en, const float* __restrict__ hw,
                          float* __restrict__ agg) {
    size_t t = (size_t)blockIdx.x * blockDim.x + threadIdx.x;  // E*32 exactly
    int e = (int)(t >> 5), c = (int)(t & 31);
    int s = src[e], d = dst[e];
    float w = en[e];
    const float4 v = *(const float4*)(hw + (size_t)s * HDIM + c * 4);
    float* p = agg + (size_t)d * HDIM + c * 4;
    unsafeAtomicAdd(p + 0, v.x * w);
    unsafeAtomicAdd(p + 1, v.y * w);
    unsafeAtomicAdd(p + 2, v.z * w);
    unsafeAtomicAdd(p + 3, v.w * w);
}

// ---------------------------------------------------------------------------
// BatchNorm (batch statistics over node axis) + ReLU, bias folded in
// ---------------------------------------------------------------------------
__global__ void zstat_k(float* __restrict__ sum, float* __restrict__ sumsq) {
    int i = threadIdx.x;
    if (i < HDIM) { sum[i] = 0.0f; sumsq[i] = 0.0f; }
}

__global__ void __launch_bounds__(256)
bnstat_k(const float* __restrict__ agg, const float* __restrict__ b,
         float* __restrict__ sum, float* __restrict__ sumsq) {
    int f  = threadIdx.x & 127;
    int rg = threadIdx.x >> 7;                 // 2 row groups per block
    float bf = b[f];
    float s = 0.0f, sq = 0.0f;
    for (int n = blockIdx.x * 2 + rg; n < N_NODES; n += gridDim.x * 2) {
        float x = agg[(size_t)n * HDIM + f] + bf;
        s += x; sq += x * x;
    }
    __shared__ float shs[2][HDIM];
    __shared__ float shq[2][HDIM];
    shs[rg][f] = s; shq[rg][f] = sq;
    __syncthreads();
    if (rg == 0) {
        unsafeAtomicAdd(&sum[f],   shs[0][f] + shs[1][f]);
        unsafeAtomicAdd(&sumsq[f], shq[0][f] + shq[1][f]);
    }
}

__global__ void bnfin_k(const float* __restrict__ sum, const float* __restrict__ sumsq,
                        const float* __restrict__ b, const float* __restrict__ gamma,
                        const float* __restrict__ beta,
                        float* __restrict__ scale, float* __restrict__ shift) {
    int f = threadIdx.x;
    const float invN = 1.0f / (float)N_NODES;
    float mu  = sum[f] * invN;
    float var = sumsq[f] * invN - mu * mu;
    float sc  = gamma[f] * rsqrtf(var + BN_EPS);
    scale[f] = sc;
    shift[f] = (b[f] - mu) * sc + beta[f];
}

__global__ void nrelu_k(const float* __restrict__ agg, const float* __restrict__ scale,
                        const float* __restrict__ shift, float* __restrict__ hout) {
    int t = blockIdx.x * blockDim.x + threadIdx.x;   // N*32 exactly
    int n = t >> 5, c = t & 31;
    float4 v  = *(const float4*)(agg   + (size_t)n * HDIM + c * 4);
    float4 sc = *(const float4*)(scale + c * 4);
    float4 sh = *(const float4*)(shift + c * 4);
    v.x = fmaxf(v.x * sc.x + sh.x, 0.0f);
    v.y = fmaxf(v.y * sc.y + sh.y, 0.0f);
    v.z = fmaxf(v.z * sc.z + sh.z, 0.0f);
    v.w = fmaxf(v.w * sc.w + sh.w, 0.0f);
    *(float4*)(hout + (size_t)n * HDIM + c * 4) = v;
}

// ---------------------------------------------------------------------------
// Global mean pool + classifier
// ---------------------------------------------------------------------------
__global__ void pzero_k(float* __restrict__ psum, float* __restrict__ pcnt) {
    int i = blockIdx.x * blockDim.x + threadIdx.x;
    if (i < NGRAPH * HDIM) psum[i] = 0.0f;
    if (i < NGRAPH)        pcnt[i] = 0.0f;
}

__global__ void pool_k(const float* __restrict__ h, const int* __restrict__ batch,
                       float* __restrict__ psum, float* __restrict__ pcnt) {
    int t = blockIdx.x * blockDim.x + threadIdx.x;   // N*32 exactly
    int n = t >> 5, c = t & 31;
    int g = batch[n];
    const float4 v = *(const float4*)(h + (size_t)n * HDIM + c * 4);
    float* p = psum + (size_t)g * HDIM + c * 4;
    unsafeAtomicAdd(p + 0, v.x);
    unsafeAtomicAdd(p + 1, v.y);
    unsafeAtomicAdd(p + 2, v.z);
    unsafeAtomicAdd(p + 3, v.w);
    if (c == 0) unsafeAtomicAdd(&pcnt[g], 1.0f);
}

__global__ void __launch_bounds__(128)
classify_k(const float* __restrict__ psum, const float* __restrict__ pcnt,
           const float* __restrict__ cw, const float* __restrict__ cb,
           float* __restrict__ out) {
    int g = blockIdx.x, t = threadIdx.x;             // 128 threads
    float inv = 1.0f / fmaxf(pcnt[g], 1.0f);
    float v = psum[(size_t)g * HDIM + t] * inv;      // pooled feature t
    __shared__ float red[HDIM];
    for (int c = 0; c < NCLS; ++c) {
        red[t] = v * cw[t * NCLS + c];
        __syncthreads();
        for (int s = 64; s > 0; s >>= 1) {
            if (t < s) red[t] += red[t + s];
            __syncthreads();
        }
        if (t == 0) out[g * NCLS + c] = red[0] + cb[c];
        __syncthreads();
    }
}

// ---------------------------------------------------------------------------
// Host-side orchestration
// ---------------------------------------------------------------------------
extern "C" void kernel_launch(void* const* d_in, const int* in_sizes, int n_in,
                              void* d_out, int out_size, void* d_ws, size_t ws_size,
                              hipStream_t stream) {
    const float* x      = (const float*)d_in[0];            // [N,128]
    const int*   ei     = (const int*)  d_in[1];            // [2,E]
    const int*   batch  = (const int*)  d_in[2];            // [N]
    const float* Ws     = (const float*)d_in[3];            // [L,128,128]
    const float* bs     = (const float*)d_in[4];            // [L,128]
    const float* gammas = (const float*)d_in[5];            // [L,128]
    const float* betas  = (const float*)d_in[6];            // [L,128]
    const float* cw     = (const float*)d_in[7];            // [128,3]
    const float* cb     = (const float*)d_in[8];            // [3]
    float*       out    = (float*)d_out;                    // [G,3]

    const int* src = ei;
    const int* dst = ei + N_EDGES;

    // Workspace carve (floats); all offsets are multiples of 4 (16B aligned)
    const size_t NH = (size_t)N_NODES * HDIM;
    float* ws      = (float*)d_ws;
    float* f_hw    = ws;                       // [N,128]
    float* f_agg   = ws + NH;                  // [N,128]
    float* f_h     = ws + 2 * NH;              // [N,128]
    float* f_en    = ws + 3 * NH;              // [E]
    float* f_dinv  = f_en + N_EDGES;           // [N]  (also used as deg)
    float* f_sum   = f_dinv + N_NODES;         // [128]
    float* f_sumsq = f_sum + HDIM;             // [128]
    float* f_scale = f_sumsq + HDIM;           // [128]
    float* f_shift = f_scale + HDIM;           // [128]
    float* f_psum  = f_shift + HDIM;           // [G,128]
    float* f_pcnt  = f_psum + (size_t)NGRAPH * HDIM;   // [G]

    const int TB = 256;
    const int nodeBlocks = (N_NODES + TB - 1) / TB;          // 196
    const int edgeBlocks = (N_EDGES + TB - 1) / TB;          // 3125
    const int nfBlocks   = (N_NODES * 32) / TB;              // 6250 (exact)
    const int efBlocks   = (N_EDGES * 32) / TB;              // 100000 (exact)
    const int gemmBlocks = N_NODES / 16;                     // 3125 (exact)

    // Normalization precompute (shared by all 3 layers)
    deg_init_k<<<nodeBlocks, TB, 0, stream>>>(f_dinv);
    deg_edge_k<<<edgeBlocks, TB, 0, stream>>>(dst, f_dinv);
    dinv_k    <<<nodeBlocks, TB, 0, stream>>>(f_dinv);
    enorm_k   <<<edgeBlocks, TB, 0, stream>>>(src, dst, f_dinv, f_en);

    const float* hin = x;
    for (int l = 0; l < NLAYERS; ++l) {
        const float* Wl = Ws + (size_t)l * HDIM * HDIM;
        gemm_wmma_k<<<gemmBlocks, TB, 0, stream>>>(hin, Wl, f_hw);
        selfloop_k <<<nfBlocks,   TB, 0, stream>>>(f_hw, f_dinv, f_agg);
        scatter_k  <<<efBlocks,   TB, 0, stream>>>(src, dst, f_en, f_hw, f_agg);
        zstat_k    <<<1, HDIM, 0, stream>>>(f_sum, f_sumsq);
        bnstat_k   <<<256, TB, 0, stream>>>(f_agg, bs + l * HDIM, f_sum, f_sumsq);
        bnfin_k    <<<1, HDIM, 0, stream>>>(f_sum, f_sumsq, bs + l * HDIM,
                                            gammas + l * HDIM, betas + l * HDIM,
                                            f_scale, f_shift);
        nrelu_k    <<<nfBlocks, TB, 0, stream>>>(f_agg, f_scale, f_shift, f_h);
        hin = f_h;
    }

    pzero_k   <<<(NGRAPH * HDIM) / TB, TB, 0, stream>>>(f_psum, f_pcnt);
    pool_k    <<<nfBlocks, TB, 0, stream>>>(f_h, batch, f_psum, f_pcnt);
    classify_k<<<NGRAPH, HDIM, 0, stream>>>(f_psum, f_pcnt, cw, cb, out);
}